// Attention_basic_28991029248450
// MI455X (gfx1250) — compile-verified
//
#include <hip/hip_runtime.h>

// ---------------------------------------------------------------------------
// Problem: B=16, S=4096.
//   q/k/v = x @ W.T + b      (3 GEMMs, M=16, N=K=4096, fp32)
//   out[b,i] = softmax_j<=i(q[b,i]*k[b,j]) . v[b,:]
//
// GEMMs are HBM-bandwidth bound (192 MB of fp32 weights, ~8.2 us at 23.3 TB/s)
// -> native fp32 WMMA (V_WMMA_F32_16X16X4_F32), NT loads for weights so x
//    stays L2-resident, K split 4-way (3072 waves) for latency hiding with
//    f32 global atomics into bias-pre-initialized accumulators.
// Attention is fused per-row online softmax (max pre-pass + single-exp pass),
// k/v staged in LDS (32 KB of the 320 KB WGP LDS).
// ---------------------------------------------------------------------------

typedef float v2f __attribute__((ext_vector_type(2)));
typedef float v8f __attribute__((ext_vector_type(8)));

#define SDIM 4096
#define BDIM 16
#define KSPLIT 4
#define NTILES (SDIM / 16)                  // 256 N-tiles per matrix
#define WAVES_TOTAL (3 * NTILES * KSPLIT)   // 3072 waves
#define KITERS ((SDIM / KSPLIT) / 4)        // 256 WMMAs per wave

// --- Kernel 0: initialize q/k/v accumulators with broadcast bias ------------
__global__ void qkv_bias_init(const float* __restrict__ bq,
                              const float* __restrict__ bk,
                              const float* __restrict__ bv,
                              float* __restrict__ qkv) {
  int tid = blockIdx.x * blockDim.x + threadIdx.x;
  if (tid >= 3 * BDIM * SDIM) return;
  int m   = tid / (BDIM * SDIM);
  int col = tid % SDIM;
  const float* bias = (m == 0) ? bq : (m == 1) ? bk : bv;
  qkv[tid] = bias[col];
}

// --- Kernel 1: fused QKV GEMM with fp32 WMMA --------------------------------
// One wave computes a 16(batch) x 16(outfeat) tile over a K/4 chunk and
// atomically accumulates into the bias-initialized buffer.
// Compile-time-constant trip count -> scalar loop control (no per-lane v_cmp,
// EXEC stays all-1s around the WMMAs as the ISA requires).
__global__ void qkv_gemm_wmma(const float* __restrict__ x,
                              const float* __restrict__ Wq,
                              const float* __restrict__ Wk,
                              const float* __restrict__ Wv,
                              float* __restrict__ qkv) {
  const int wave = (blockIdx.x * blockDim.x + threadIdx.x) >> 5;
  const int lane = threadIdx.x & 31;

  const int mtx    = wave / (NTILES * KSPLIT);
  const int rem    = wave % (NTILES * KSPLIT);
  const int ntile  = rem / KSPLIT;
  const int kchunk = rem % KSPLIT;

  const float* __restrict__ W = (mtx == 0) ? Wq : (mtx == 1) ? Wk : Wv;
  float* __restrict__ out = qkv + mtx * (BDIM * SDIM);

  const int n   = ntile * 16;
  const int l16 = lane & 15;
  const int hh  = lane >> 4;          // 0: K pair (k,k+1); 1: K pair (k+2,k+3)
  const int k0  = kchunk * (SDIM / KSPLIT);

  // A fragment (x tile, 16x4):  lane L holds x[L&15][k + 2*hh .. +1]
  // B fragment (W.T tile, 4x16): lane L holds W[n + (L&15)][k + 2*hh .. +1]
  const float* pA = x + l16 * SDIM + 2 * hh + k0;
  const float* pB = W + (n + l16) * SDIM + 2 * hh + k0;

  v8f c = {};
#pragma unroll 8
  for (int it = 0; it < KITERS; ++it) {
    v2f a = *(const v2f*)pA;
    // Non-temporal: weights are streamed exactly once; keep x hot in L2.
    v2f b = __builtin_nontemporal_load((const v2f*)pB);
    c = __builtin_amdgcn_wmma_f32_16x16x4_f32(
        /*neg_a=*/false, a, /*neg_b=*/false, b,
        /*c_mod=*/(short)0, c, /*reuse_a=*/false, /*reuse_b=*/false);
    pA += 4;
    pB += 4;
  }

  // C/D layout: VGPR v, lanes 0-15 -> row v, lanes 16-31 -> row v+8; col = L&15
#pragma unroll
  for (int v = 0; v < 8; ++v) {
    int row = v + 8 * hh;
    unsafeAtomicAdd(&out[row * SDIM + n + l16], c[v]);
  }
}

// --- Kernel 2: causal d=1 attention (outer-product scores, online softmax) --
// Block = 256 threads handling 256 consecutive rows i for batch b = blockIdx.y.
// k[b,:], v[b,:] staged in LDS (2 * 16 KB).
__global__ void attn_softmax_kernel(const float* __restrict__ qkv,
                                    float* __restrict__ out) {
  __shared__ float ks[SDIM];
  __shared__ float vs[SDIM];

  const int b   = blockIdx.y;
  const int tid = threadIdx.x;
  const float* __restrict__ q = qkv;
  const float* __restrict__ k = qkv + BDIM * SDIM;
  const float* __restrict__ v = qkv + 2 * BDIM * SDIM;

  for (int j = tid; j < SDIM; j += blockDim.x) {
    ks[j] = k[b * SDIM + j];
    vs[j] = v[b * SDIM + j];
  }
  __syncthreads();

  const int i = blockIdx.x * blockDim.x + tid;
  const float qi = q[b * SDIM + i];

  // Pass 1: row max (cheap VALU only, avoids a second exp in the main loop).
  float m = -3.402823466e38f;
  for (int j = 0; j <= i; ++j) m = fmaxf(m, qi * ks[j]);

  // Pass 2: single-exp softmax accumulation. Masked (j>i) entries contribute
  // exp(-1e9 - m) ~ 0, and every row has >=1 unmasked entry, so skipping them
  // is exact.
  float s = 0.0f, o = 0.0f;
  for (int j = 0; j <= i; ++j) {
    float p = __expf(qi * ks[j] - m);
    s += p;
    o = fmaf(p, vs[j], o);
  }
  out[b * SDIM + i] = o / s;
}

// ---------------------------------------------------------------------------
extern "C" void kernel_launch(void* const* d_in, const int* in_sizes, int n_in,
                              void* d_out, int out_size, void* d_ws, size_t ws_size,
                              hipStream_t stream) {
  const float* x  = (const float*)d_in[0];
  const float* Wq = (const float*)d_in[1];
  const float* bq = (const float*)d_in[2];
  const float* Wk = (const float*)d_in[3];
  const float* bk = (const float*)d_in[4];
  const float* Wv = (const float*)d_in[5];
  const float* bv = (const float*)d_in[6];
  float* out = (float*)d_out;
  float* qkv = (float*)d_ws;   // 3 * 16 * 4096 floats = 768 KB scratch

  // 0) bias init of q/k/v accumulators
  {
    int total = 3 * BDIM * SDIM;
    qkv_bias_init<<<(total + 255) / 256, 256, 0, stream>>>(bq, bk, bv, qkv);
  }
  // 1) fused QKV GEMM: 3072 waves (3 matrices x 256 N-tiles x 4 K-chunks)
  {
    int threads = WAVES_TOTAL * 32;
    qkv_gemm_wmma<<<threads / 256, 256, 0, stream>>>(x, Wq, Wk, Wv, qkv);
  }
  // 2) fused causal softmax-attention
  {
    dim3 grid(SDIM / 256, BDIM);
    attn_softmax_kernel<<<grid, 256, 0, stream>>>(qkv, out);
  }
}